// VectorQuantizer_31602369364090
// MI455X (gfx1250) — compile-verified
//
#include <hip/hip_runtime.h>
#include <stdint.h>

// VQ argmin via fused fp32 WMMA GEMM + in-register argmax (gfx1250 / MI455X).
// Avoids materializing the 1 GB [T,N] distance matrix entirely.
// - fp32 16x16x4 WMMA keeps argmin bit-faithful to the fp32 reference.
// - Two independent accumulator chains per iteration keep the XDL pipe fed.
// - Global->LDS staging is software-pipelined one chunk ahead to hide L2 latency.
// - Argmax update: 1 v_cmp + 2 dual-issued v_cndmask per row (strict '>' with
//   ascending n reproduces jnp.argmin's first-occurrence tie-break exactly).

typedef float f32x2 __attribute__((ext_vector_type(2)));
typedef float f32x4 __attribute__((ext_vector_type(4)));
typedef float f32x8 __attribute__((ext_vector_type(8)));

#define N_E        8192
#define E_DIM      32
#define T_TOK      32768
#define CHUNK      128        // codes staged in LDS per iteration (18 KB padded)
#define LDS_STRIDE 36         // floats/code row: 16B-aligned, conflict-free for b64 reads
#define WAVES      8
#define BLOCK      (WAVES * 32)
#define TOK_PER_BLOCK (WAVES * 16)
#define STAGE_V4   ((CHUNK * E_DIM) / (BLOCK * 4))   // float4 per thread per chunk = 4

// ---------------- kernel 1: normalize codebook rows, zero loss slot ----------
__global__ __launch_bounds__(256)
void vq_prep_emb(const float* __restrict__ emb, float* __restrict__ en,
                 float* __restrict__ loss_out) {
    if (blockIdx.x == 0 && threadIdx.x == 0) *loss_out = 0.0f;
    int tid = blockIdx.x * 256 + threadIdx.x;
    int row = tid >> 3;                 // 8 lanes per code row
    int l8  = tid & 7;
    if (row >= N_E) return;
    f32x4 v = *(const f32x4*)(emb + row * E_DIM + l8 * 4);
    float ss = v.x * v.x + v.y * v.y + v.z * v.z + v.w * v.w;
    ss += __shfl_xor(ss, 1);
    ss += __shfl_xor(ss, 2);
    ss += __shfl_xor(ss, 4);            // xor<8 stays inside the 8-lane group
    float inv = 1.0f / fmaxf(sqrtf(ss), 1e-12f);
    v.x *= inv; v.y *= inv; v.z *= inv; v.w *= inv;
    *(f32x4*)(en + row * E_DIM + l8 * 4) = v;
}

// ---------------- kernel 2: fused normalize(z) + GEMM-argmax + gather -------
__global__ __launch_bounds__(BLOCK)
void vq_main(const float* __restrict__ z, const float* __restrict__ en,
             float* __restrict__ zq_out, float* __restrict__ loss_out,
             float* __restrict__ idx_out) {
    __shared__ float en_lds[CHUNK * LDS_STRIDE];
    __shared__ int   idx_lds[WAVES * 16];

    const int tid  = threadIdx.x;
    const int wave = tid >> 5;
    const int lane = tid & 31;
    const int half = lane >> 4;          // 0: local K 0,1 | 1: local K 2,3
    const int l16  = lane & 15;

    const int t0  = blockIdx.x * TOK_PER_BLOCK + wave * 16;
    const int row = t0 + l16;            // token row co-owned by lane pair (l, l+16)

    // per-thread staging addresses (fixed across chunks)
    const int s_lin = tid;                            // float4 index base
    int s_r[STAGE_V4], s_c4[STAGE_V4];
#pragma unroll
    for (int j = 0; j < STAGE_V4; ++j) {
        int lin = s_lin + j * BLOCK;
        s_r[j]  = lin >> 3;                           // 8 float4 per code
        s_c4[j] = lin & 7;
    }

    // ---- load this wave's 16x32 z tile in WMMA A layout and l2-normalize ----
    f32x2 a[8];
    float ss = 0.0f;
#pragma unroll
    for (int kk = 0; kk < 8; ++kk) {
        a[kk] = *(const f32x2*)(z + row * E_DIM + 4 * kk + 2 * half);
        ss += a[kk].x * a[kk].x + a[kk].y * a[kk].y;
    }
    ss += __shfl_xor(ss, 16);            // lane pair sums the full 32-elem row
    float inv = 1.0f / fmaxf(sqrtf(ss), 1e-12f);
#pragma unroll
    for (int kk = 0; kk < 8; ++kk) { a[kk].x *= inv; a[kk].y *= inv; }

    float bval[8];
    int   bidx[8];
#pragma unroll
    for (int v = 0; v < 8; ++v) { bval[v] = -3.402823466e38f; bidx[v] = 0; }

    // ---- prefetch chunk 0 into registers ------------------------------------
    f32x4 stage[STAGE_V4];
#pragma unroll
    for (int j = 0; j < STAGE_V4; ++j)
        stage[j] = *(const f32x4*)(en + s_r[j] * E_DIM + s_c4[j] * 4);

    // ---- sweep all codes in LDS-staged chunks, WMMA + fused argmax ----------
    for (int cbase = 0; cbase < N_E; cbase += CHUNK) {
        __syncthreads();                 // previous chunk's LDS fully consumed
#pragma unroll
        for (int j = 0; j < STAGE_V4; ++j)
            *(f32x4*)(&en_lds[s_r[j] * LDS_STRIDE + s_c4[j] * 4]) = stage[j];
        __syncthreads();

        // prefetch next chunk while this one is being consumed (hides L2 latency)
        if (cbase + CHUNK < N_E) {
#pragma unroll
            for (int j = 0; j < STAGE_V4; ++j)
                stage[j] = *(const f32x4*)(en + (cbase + CHUNK + s_r[j]) * E_DIM
                                              + s_c4[j] * 4);
        }

#pragma unroll
        for (int sub = 0; sub < CHUNK / 16; sub += 2) {
            // two independent accumulator chains -> XDL pipe never starves
            f32x8 c0 = {0.f, 0.f, 0.f, 0.f, 0.f, 0.f, 0.f, 0.f};
            f32x8 c1 = {0.f, 0.f, 0.f, 0.f, 0.f, 0.f, 0.f, 0.f};
            const float* bp0 = &en_lds[(sub * 16 + l16) * LDS_STRIDE + 2 * half];
            const float* bp1 = bp0 + 16 * LDS_STRIDE;
#pragma unroll
            for (int kk = 0; kk < 8; ++kk) {
                f32x2 b0 = *(const f32x2*)(bp0 + 4 * kk);   // ds_load_b64, all-64-bank
                f32x2 b1 = *(const f32x2*)(bp1 + 4 * kk);
                c0 = __builtin_amdgcn_wmma_f32_16x16x4_f32(
                        false, a[kk], false, b0, (short)0, c0, false, false);
                c1 = __builtin_amdgcn_wmma_f32_16x16x4_f32(
                        false, a[kk], false, b1, (short)0, c1, false, false);
            }
            // C layout: VGPR v -> (lane<16 ? M=v, N=lane : M=v+8, N=lane-16)
            int n0 = cbase + sub * 16 + l16;
            int n1 = n0 + 16;
#pragma unroll
            for (int v = 0; v < 8; ++v) {
                bool g0 = c0[v] > bval[v];          // strict > => first-occurrence ties
                bidx[v] = g0 ? n0 : bidx[v];        // v_dual_cndmask (shares VCC)
                bval[v] = g0 ? c0[v] : bval[v];
                bool g1 = c1[v] > bval[v];
                bidx[v] = g1 ? n1 : bidx[v];
                bval[v] = g1 ? c1[v] : bval[v];
            }
        }
    }

    // ---- butterfly argmax across the 16 lanes sharing each row --------------
#pragma unroll
    for (int m = 8; m >= 1; m >>= 1) {
#pragma unroll
        for (int v = 0; v < 8; ++v) {
            float ov = __shfl_xor(bval[v], m);      // xor<16: stays in 16-lane group
            int   oi = __shfl_xor(bidx[v], m);
            bool take = (ov > bval[v]) | ((ov == bval[v]) & (oi < bidx[v]));
            if (take) { bval[v] = ov; bidx[v] = oi; }
        }
    }
    if (l16 == 0) {                       // lane 0 -> rows 0..7, lane 16 -> rows 8..15
#pragma unroll
        for (int v = 0; v < 8; ++v)
            idx_lds[wave * 16 + half * 8 + v] = bidx[v];
    }
    __syncthreads();

    // ---- gather en[idx], straight-through output, fused loss ----------------
    int myidx = idx_lds[wave * 16 + l16];
    float lsum = 0.0f;
#pragma unroll
    for (int kk = 0; kk < 8; ++kk) {
        int col = 4 * kk + 2 * half;
        f32x2 e = *(const f32x2*)(en + myidx * E_DIM + col);
        float d0 = e.x - a[kk].x;
        float d1 = e.y - a[kk].y;
        f32x2 o; o.x = a[kk].x + d0; o.y = a[kk].y + d1;   // z + (z_q - z), as ref
        *(f32x2*)(zq_out + row * E_DIM + col) = o;
        lsum += d0 * d0 + d1 * d1;
    }
    if (half == 0) idx_out[row] = (float)myidx;

    lsum += __shfl_xor(lsum, 1);
    lsum += __shfl_xor(lsum, 2);
    lsum += __shfl_xor(lsum, 4);
    lsum += __shfl_xor(lsum, 8);
    lsum += __shfl_xor(lsum, 16);
    if (lane == 0)
        atomicAdd(loss_out, lsum * (1.25f / 1048576.0f));  // (1+BETA)/numel
}

// ---------------- launch -----------------------------------------------------
extern "C" void kernel_launch(void* const* d_in, const int* in_sizes, int n_in,
                              void* d_out, int out_size, void* d_ws, size_t ws_size,
                              hipStream_t stream) {
    (void)in_sizes; (void)n_in; (void)out_size; (void)ws_size;
    const float* zin = (const float*)d_in[0];   // [8,64,64,32] fp32
    const float* emb = (const float*)d_in[1];   // [8192,32]    fp32
    float* out  = (float*)d_out;                // [zq | loss | idx] concatenated
    float* zq   = out;
    float* loss = out + (size_t)T_TOK * E_DIM;
    float* idxo = out + (size_t)T_TOK * E_DIM + 1;
    float* en   = (float*)d_ws;                 // normalized codebook, 1 MB

    vq_prep_emb<<<N_E / 32, 256, 0, stream>>>(emb, en, loss);
    vq_main<<<T_TOK / TOK_PER_BLOCK, BLOCK, 0, stream>>>(zin, en, zq, loss, idxo);
}